// LinkPredictionModel_4964982194352
// MI455X (gfx1250) — compile-verified
//
#include <hip/hip_runtime.h>
#include <hip/hip_bf16.h>

// Link-prediction GNN (3x SAGE mean-aggr + 2-layer MLP decoder), fp32.
// Memory-bound: scatter/gather dominates; GEMMs use native-f32 WMMA
// (V_WMMA_F32_16X16X4_F32). K=256 loop split into two branch-free halves;
// mean-divide folded into a one-time accumulator rescale (linearity).

#define DH 128            // hidden dim
#define DK 256            // concat K dim = 2*DH

typedef __attribute__((ext_vector_type(2))) float v2f;
typedef __attribute__((ext_vector_type(8))) float v8f;

__global__ void zero_f32(float* __restrict__ p, long n) {
    long i = (long)blockIdx.x * blockDim.x + threadIdx.x;
    if (i < n) p[i] = 0.0f;
}

// One wave (32 lanes) per edge; each lane moves 4 floats (float4 load + 4 f32 atomics).
__global__ void __launch_bounds__(256)
scatter_add(const float* __restrict__ x, const int* __restrict__ src,
            const int* __restrict__ dst, float* __restrict__ agg,
            float* __restrict__ cnt /* null after layer 1 */, int n_edges) {
    long t = (long)blockIdx.x * blockDim.x + threadIdx.x;
    int e = (int)(t >> 5);
    if (e >= n_edges) return;
    int lane = (int)(t & 31);
    int s = src[e], d = dst[e];
    float4 v = ((const float4*)(x + (long)s * DH))[lane];
    float* ap = agg + (long)d * DH + lane * 4;
    atomicAdd(ap + 0, v.x);
    atomicAdd(ap + 1, v.y);
    atomicAdd(ap + 2, v.z);
    atomicAdd(ap + 3, v.w);
    if (cnt != nullptr && lane == 0) atomicAdd(cnt + d, 1.0f);
}

// Branch-free WMMA half-GEMM: acc += Arows @ W over K=DH, step 4.
// Arow per lane: row = lane&15, K pair = 4*it + 2*(lane>>4).
__device__ __forceinline__ void
wmma_half(const float* __restrict__ arow, const float* __restrict__ W,
          int kh, int col0, int col1, v8f& acc0, v8f& acc1) {
#pragma unroll 8
    for (int it = 0; it < DH / 4; ++it) {
        const int kb = it * 4 + kh * 2;
        v2f a;
        a.x = arow[kb];
        a.y = arow[kb + 1];
        const float* __restrict__ Wk = W + (long)kb * DH;
        v2f b0, b1;
        b0.x = Wk[col0]; b0.y = Wk[DH + col0];
        b1.x = Wk[col1]; b1.y = Wk[DH + col1];
        acc0 = __builtin_amdgcn_wmma_f32_16x16x4_f32(false, a, false, b0,
                                                     (short)0, acc0, false, false);
        acc1 = __builtin_amdgcn_wmma_f32_16x16x4_f32(false, a, false, b1,
                                                     (short)0, acc1, false, false);
    }
}

// Fused SAGE layer: out = act( (agg/max(cnt,1)) @ Wl + bl + xin @ Wr )
// Block = 128 threads = 4 waves; block owns 16 rows x 128 cols; each wave
// owns two 16x16 N-tiles. agg@Wl accumulated raw, then rescaled per C-row
// by 1/max(cnt,1) (linearity), then xin@Wr accumulated on top.
__global__ void __launch_bounds__(128)
sage_wmma(const float* __restrict__ agg, const float* __restrict__ cnt,
          const float* __restrict__ xin, const float* __restrict__ Wl,
          const float* __restrict__ bl, const float* __restrict__ Wr,
          float* __restrict__ out, int n_nodes, int do_relu) {
    const int lane = threadIdx.x & 31;
    const int wave = threadIdx.x >> 5;
    const int r    = lane & 15;       // A row-in-block / B,C col-in-tile
    const int kh   = lane >> 4;       // K sub-pair select / C row-half select
    const int col0 = wave * 32 + r;
    const int col1 = col0 + 16;
    const int base = blockIdx.x * 16;
    int grow = base + r;
    int gl = grow < n_nodes ? grow : n_nodes - 1;   // clamp loads, keep EXEC full

    const float* __restrict__ arow = agg + (long)gl * DH;
    const float* __restrict__ xrow = xin + (long)gl * DH;

    v8f acc0 = {}; v8f acc1 = {};

    // ---- Half 1: raw neighbor-sum @ Wl
    wmma_half(arow, Wl, kh, col0, col1, acc0, acc1);

    // ---- Rescale accumulators by per-row 1/max(deg,1): C row m = vg + 8*kh
#pragma unroll
    for (int vg = 0; vg < 8; ++vg) {
        int m  = base + vg + kh * 8;
        int mc = m < n_nodes ? m : n_nodes - 1;
        float ic = 1.0f / fmaxf(cnt[mc], 1.0f);
        acc0[vg] *= ic;
        acc1[vg] *= ic;
    }

    // ---- Half 2: self features @ Wr
    wmma_half(xrow, Wr, kh, col0, col1, acc0, acc1);

    const float bias0 = bl[col0];
    const float bias1 = bl[col1];
#pragma unroll
    for (int vg = 0; vg < 8; ++vg) {
        int m = vg + kh * 8;
        float o0 = acc0[vg] + bias0;
        float o1 = acc1[vg] + bias1;
        if (do_relu) { o0 = fmaxf(o0, 0.0f); o1 = fmaxf(o1, 0.0f); }
        if (base + m < n_nodes) {
            out[(long)(base + m) * DH + col0] = o0;
            out[(long)(base + m) * DH + col1] = o1;
        }
    }
}

// Fully fused decoder: for 16 label pairs per block,
//   hdec = relu([z[row] | z[col]] @ Wd1 + bd1); out = hdec @ Wd2 + bd2
// Two branch-free WMMA half-loops (z[row] @ Wd1[:128], z[col] @ Wd1[128:]),
// then per-lane dot with Wd2, shfl-xor reduce over each 16-lane half
// (disjoint C row halves), LDS combine across the 4 waves. No hdec in HBM.
__global__ void __launch_bounds__(128)
decoder_wmma(const float* __restrict__ z, const int* __restrict__ eli,
             const float* __restrict__ Wd1, const float* __restrict__ bd1,
             const float* __restrict__ Wd2, const float* __restrict__ bd2,
             float* __restrict__ out, int n_label) {
    __shared__ float lds_p[4][16];
    const int lane = threadIdx.x & 31;
    const int wave = threadIdx.x >> 5;
    const int r    = lane & 15;
    const int kh   = lane >> 4;
    const int col0 = wave * 32 + r;
    const int col1 = col0 + 16;
    const int base = blockIdx.x * 16;
    int e  = base + r;
    int el = e < n_label ? e : n_label - 1;
    const int ri = eli[el];
    const int ci = eli[n_label + el];
    const float* __restrict__ zr = z + (long)ri * DH;
    const float* __restrict__ zc = z + (long)ci * DH;

    v8f acc0 = {}; v8f acc1 = {};
    wmma_half(zr, Wd1,                 kh, col0, col1, acc0, acc1);
    wmma_half(zc, Wd1 + (long)DH * DH, kh, col0, col1, acc0, acc1);

    const float bias0 = bd1[col0], bias1 = bd1[col1];
    const float w20 = Wd2[col0],   w21 = Wd2[col1];
    float p[8];
#pragma unroll
    for (int vg = 0; vg < 8; ++vg) {
        float h0 = fmaxf(acc0[vg] + bias0, 0.0f);
        float h1 = fmaxf(acc1[vg] + bias1, 0.0f);
        p[vg] = h0 * w20 + h1 * w21;
    }
    // Reduce over the 16 lanes sharing a row half (masks 1..8 keep halves apart).
#pragma unroll
    for (int off = 1; off < 16; off <<= 1)
#pragma unroll
        for (int vg = 0; vg < 8; ++vg)
            p[vg] += __shfl_xor(p[vg], off, 32);

    if (r == 0) {                        // lanes 0 (rows 0..7) and 16 (rows 8..15)
#pragma unroll
        for (int vg = 0; vg < 8; ++vg) lds_p[wave][kh * 8 + vg] = p[vg];
    }
    __syncthreads();
    if (threadIdx.x < 16) {
        int oe = base + threadIdx.x;
        if (oe < n_label) {
            out[oe] = lds_p[0][threadIdx.x] + lds_p[1][threadIdx.x] +
                      lds_p[2][threadIdx.x] + lds_p[3][threadIdx.x] + bd2[0];
        }
    }
}

extern "C" void kernel_launch(void* const* d_in, const int* in_sizes, int n_in,
                              void* d_out, int out_size, void* d_ws, size_t ws_size,
                              hipStream_t stream) {
    const float* x   = (const float*)d_in[0];
    const int*   ei  = (const int*)d_in[1];
    const int*   eli = (const int*)d_in[2];
    const float* W1l = (const float*)d_in[3];
    const float* b1l = (const float*)d_in[4];
    const float* W1r = (const float*)d_in[5];
    const float* W2l = (const float*)d_in[6];
    const float* b2l = (const float*)d_in[7];
    const float* W2r = (const float*)d_in[8];
    const float* W3l = (const float*)d_in[9];
    const float* b3l = (const float*)d_in[10];
    const float* W3r = (const float*)d_in[11];
    const float* Wd1 = (const float*)d_in[12];
    const float* bd1 = (const float*)d_in[13];
    const float* Wd2 = (const float*)d_in[14];
    const float* bd2 = (const float*)d_in[15];
    float* out = (float*)d_out;

    const int N = in_sizes[0] / DH;     // 50000 nodes
    const int E = in_sizes[1] / 2;      // 800000 edges
    const int L = in_sizes[2] / 2;      // 200000 label pairs

    // Workspace: agg[N*DH] | cnt[N] | hA[N*DH] | hB[N*DH]  (~77 MB)
    float* agg = (float*)d_ws;
    float* cnt = agg + (long)N * DH;
    float* hA  = cnt + N;
    float* hB  = hA + (long)N * DH;

    const int* src = ei;
    const int* dst = ei + E;

    const long aggN = (long)N * DH;
    const int zthr  = 256;
    const dim3 zgridA((unsigned)((aggN + N + zthr - 1) / zthr));   // agg+cnt (contiguous)
    const dim3 zgrid((unsigned)((aggN + zthr - 1) / zthr));        // agg only
    const dim3 sgrid((unsigned)(((long)E * 32 + 255) / 256));
    const dim3 ngrid((unsigned)((N + 15) / 16));
    const dim3 lgrid((unsigned)((L + 15) / 16));

    // ---- Layer 1: x -> hA (ReLU); also builds degree counts (layer-invariant)
    zero_f32<<<zgridA, zthr, 0, stream>>>(agg, aggN + N);
    scatter_add<<<sgrid, 256, 0, stream>>>(x, src, dst, agg, cnt, E);
    sage_wmma<<<ngrid, 128, 0, stream>>>(agg, cnt, x, W1l, b1l, W1r, hA, N, 1);

    // ---- Layer 2: hA -> hB (ReLU)
    zero_f32<<<zgrid, zthr, 0, stream>>>(agg, aggN);
    scatter_add<<<sgrid, 256, 0, stream>>>(hA, src, dst, agg, nullptr, E);
    sage_wmma<<<ngrid, 128, 0, stream>>>(agg, cnt, hA, W2l, b2l, W2r, hB, N, 1);

    // ---- Layer 3: hB -> hA (= z, no ReLU)
    zero_f32<<<zgrid, zthr, 0, stream>>>(agg, aggN);
    scatter_add<<<sgrid, 256, 0, stream>>>(hB, src, dst, agg, nullptr, E);
    sage_wmma<<<ngrid, 128, 0, stream>>>(agg, cnt, hB, W3l, b3l, W3r, hA, N, 0);

    // ---- Fused decoder: gather pairs -> WMMA MLP -> scores
    decoder_wmma<<<lgrid, 128, 0, stream>>>(hA, eli, Wd1, bd1, Wd2, bd2, out, L);
}